// JointsMSELoss_72653666779192
// MI455X (gfx1250) — compile-verified
//
#include <hip/hip_runtime.h>
#include <math.h>

#define B_   64
#define J_   16
#define COL_ 96
#define HW_  (COL_ * COL_)   // 9216
#define BJ_  (B_ * J_)       // 1024

typedef float v2f __attribute__((ext_vector_type(2)));
typedef float v8f __attribute__((ext_vector_type(8)));

// Wave32 sum reduction via V_WMMA_F32_16X16X4_F32.
// A (16x4 f32): VGPR0 lanes0-15 = col K=0, VGPR1 lanes0-15 = K=1,
//               VGPR0 lanes16-31 = K=2, VGPR1 lanes16-31 = K=3.
// With a = {acc, 0} per lane and B = all-ones, D[m][n] = acc[m] + acc[m+16].
// D layout: lane L<16 holds D[0..7][L] in 8 VGPRs, lane L>=16 holds D[8..15][L-16].
// Sum of the 8 D regs per lane + shfl_xor(16) => full wave sum in every lane.
__device__ __forceinline__ float wave_sum_wmma(float v) {
  v2f a;    a.x = v;    a.y = 0.0f;
  v2f ones; ones.x = 1.0f; ones.y = 1.0f;
  v8f c = {0.f, 0.f, 0.f, 0.f, 0.f, 0.f, 0.f, 0.f};
  c = __builtin_amdgcn_wmma_f32_16x16x4_f32(false, a, false, ones,
                                            (short)0, c, false, false);
  float r = c[0] + c[1] + c[2] + c[3] + c[4] + c[5] + c[6] + c[7];
  r += __shfl_xor(r, 16, 32);
  return r;
}

__global__ __launch_bounds__(256) void joints_partial_kernel(
    const float* __restrict__ offset,   // [B, 2J, HW]
    const float* __restrict__ heatmap,  // [B, J, HW]
    const float* __restrict__ target,   // [B, J, HW]
    const float* __restrict__ tw,       // [B, J, 1]
    float* __restrict__ xout,           // d_out + 1 : [B, J, 2]
    float* __restrict__ partials)       // d_ws : [B*J] squared-diff sums
{
  const int bj   = blockIdx.x;          // b*J + j
  const int tid  = threadIdx.x;
  const int lane = tid & 31;
  const int wave = tid >> 5;

  const float w = tw[bj];
  const float4* __restrict__ p4 = (const float4*)(heatmap + (size_t)bj * HW_);
  const float4* __restrict__ g4 = (const float4*)(target  + (size_t)bj * HW_);

  float acc = 0.0f;
  float mx  = -3.402823466e+38f;
  int   mi  = 0x7fffffff;

  // HW = 9216 floats = 2304 float4; 256 threads * 9 iterations.
#pragma unroll
  for (int it = 0; it < 9; ++it) {
    const int k = tid + it * 256;       // 0..2303
    const float4 p = p4[k];
    const float4 g = g4[k];
    const float d0 = p.x * w - g.x * w;
    const float d1 = p.y * w - g.y * w;
    const float d2 = p.z * w - g.z * w;
    const float d3 = p.w * w - g.w * w;
    acc += d0 * d0 + d1 * d1 + d2 * d2 + d3 * d3;
    const int base = 4 * k;
    if (p.x > mx) { mx = p.x; mi = base;     }
    if (p.y > mx) { mx = p.y; mi = base + 1; }
    if (p.z > mx) { mx = p.z; mi = base + 2; }
    if (p.w > mx) { mx = p.w; mi = base + 3; }
  }

  // Wave sum of squared diffs via WMMA (exact f32 accumulation).
  const float wsum = wave_sum_wmma(acc);

  // Wave argmax; smaller-index tie-break mimics jnp.argmax first occurrence.
  for (int off = 16; off > 0; off >>= 1) {
    const float ov = __shfl_down(mx, off, 32);
    const int   oi = __shfl_down(mi, off, 32);
    if (ov > mx || (ov == mx && oi < mi)) { mx = ov; mi = oi; }
  }

  __shared__ float s_sum[8];
  __shared__ float s_max[8];
  __shared__ int   s_idx[8];
  if (lane == 0) { s_sum[wave] = wsum; s_max[wave] = mx; s_idx[wave] = mi; }
  __syncthreads();

  if (tid == 0) {
    float tot = 0.0f;
    float bmx = s_max[0];
    int   bmi = s_idx[0];
#pragma unroll
    for (int i = 0; i < 8; ++i) {
      tot += s_sum[i];
      if (i > 0 && (s_max[i] > bmx || (s_max[i] == bmx && s_idx[i] < bmi))) {
        bmx = s_max[i];
        bmi = s_idx[i];
      }
    }
    partials[bj] = tot;

    // Decode argmax -> coordinates, gather the two offset values.
    const int yc = bmi / COL_;
    const int xc = bmi - yc * COL_;
    const int b  = bj / J_;
    const int j  = bj - b * J_;
    const size_t ox_idx = ((size_t)b * (2 * J_) + j)       * HW_ + bmi;
    const size_t oy_idx = ((size_t)b * (2 * J_) + J_ + j)  * HW_ + bmi;
    const float ox = offset[ox_idx];
    const float oy = offset[oy_idx];
    xout[2 * bj + 0] = (ox + (float)xc) * (float)COL_;
    xout[2 * bj + 1] = (oy + (float)yc) * (float)COL_;
  }
}

__global__ __launch_bounds__(1024) void joints_finalize_kernel(
    const float* __restrict__ partials, // [1024]
    const float* __restrict__ xout,     // d_out + 1 : [1024, 2]
    const float* __restrict__ joints,   // [1024, 2]
    const float* __restrict__ jvis,     // [1024, 2]
    float* __restrict__ out0)           // d_out[0]
{
  const int t    = threadIdx.x;         // 0..1023, one (b,j) pair per thread
  const int lane = t & 31;
  const int wave = t >> 5;

  float s1 = partials[t];
  const float x0 = xout[2 * t],   x1 = xout[2 * t + 1];
  const float j0 = joints[2 * t], j1 = joints[2 * t + 1];
  const float v0 = jvis[2 * t],   v1 = jvis[2 * t + 1];
  const float dx = (x0 - j0) * v0 * (1.0f / 256.0f);
  const float dy = (x1 - j1) * v1 * (1.0f / 256.0f);
  float s2 = dx * dx + dy * dy;
  float sv = v0 + v1;

  for (int off = 16; off > 0; off >>= 1) {
    s1 += __shfl_xor(s1, off, 32);
    s2 += __shfl_xor(s2, off, 32);
    sv += __shfl_xor(sv, off, 32);
  }

  __shared__ float a1[32], a2[32], a3[32];
  if (lane == 0) { a1[wave] = s1; a2[wave] = s2; a3[wave] = sv; }
  __syncthreads();

  if (t == 0) {
    float S1 = 0.f, S2 = 0.f, SV = 0.f;
#pragma unroll
    for (int i = 0; i < 32; ++i) { S1 += a1[i]; S2 += a2[i]; SV += a3[i]; }
    const float d1 = 0.5f * S1 / (float)((long long)BJ_ * HW_);
    const float N2 = SV * 0.5f;
    const float d2 = 0.5f * sqrtf(S2) / N2;
    out0[0] = d1 + d2;
  }
}

extern "C" void kernel_launch(void* const* d_in, const int* in_sizes, int n_in,
                              void* d_out, int out_size, void* d_ws, size_t ws_size,
                              hipStream_t stream) {
  const float* offset  = (const float*)d_in[0];
  const float* heatmap = (const float*)d_in[1];
  const float* target  = (const float*)d_in[2];
  const float* tw      = (const float*)d_in[3];
  const float* joints  = (const float*)d_in[4];
  const float* jvis    = (const float*)d_in[5];

  float* out      = (float*)d_out;   // [0] = scalar loss, [1..2048] = x
  float* xout     = out + 1;
  float* partials = (float*)d_ws;    // 1024 floats

  joints_partial_kernel<<<BJ_, 256, 0, stream>>>(offset, heatmap, target, tw,
                                                 xout, partials);
  joints_finalize_kernel<<<1, 1024, 0, stream>>>(partials, xout, joints, jvis,
                                                 out);
}